// Window_19292993093749
// MI455X (gfx1250) — compile-verified
//
#include <hip/hip_runtime.h>

typedef __attribute__((ext_vector_type(2))) float v2f;
typedef __attribute__((ext_vector_type(8))) float v8f;

#define B_DIM 1024
#define C_DIM 1024
#define K_DIM 20
#define L_DIM 2048
#define V_DIM 80
#define NP    60          // 3*K
#define NPAD  64          // padded column count for WMMA tiles
#define XS_STRIDE 1028    // 1024 + 4: bank-conflict-free A-fragment reads

// -----------------------------------------------------------------------------
// Kernel 0: transpose + zero-pad W [1024][60] -> Wt [64][1024] in workspace.
// Makes the GEMM B-fragment a branch-free contiguous b64 load; cols 60..63
// are real zeros so no masking is ever needed.
// -----------------------------------------------------------------------------
__global__ __launch_bounds__(256)
void wpad_kernel(const float* __restrict__ W, float* __restrict__ Wt) {
  const int idx = blockIdx.x * 256 + threadIdx.x;   // over 64*1024
  const int c = idx >> 10;                          // 0..63 (output row = col)
  const int k = idx & (C_DIM - 1);                  // 0..1023
  Wt[(size_t)c * C_DIM + k] = (c < NP) ? W[(size_t)k * NP + c] : 0.0f;
}

// -----------------------------------------------------------------------------
// Kernel 1: params = exp(x @ W + b) via V_WMMA_F32_16X16X4_F32 (full f32).
// grid = 64 blocks (16-row x tile each), block = 128 threads (4 waves);
// wave w owns column tile 16*w of the padded 64-column output.
// x tile staged in LDS with one coalesced float4 sweep; inner loop is
// ds_load_b64 (A) + global_load_b64 (B, from zero-padded Wt) + wmma.
// -----------------------------------------------------------------------------
__global__ __launch_bounds__(128)
void params_gemm_kernel(const float* __restrict__ x,
                        const float* __restrict__ Wt,
                        const float* __restrict__ bias,
                        float* __restrict__ params) {
  __shared__ float xs[16 * XS_STRIDE];
  const int tid  = threadIdx.x;
  const int row0 = blockIdx.x * 16;

  // Stage 16x1024 x tile: 4096 float4, 128 threads -> 32 each, coalesced.
#pragma unroll 4
  for (int i = 0; i < 32; ++i) {
    const int idx = tid + 128 * i;     // float4 index
    const int r   = idx >> 8;          // / (1024/4)
    const int c4  = idx & 255;
    const float4 v = ((const float4*)(x + (size_t)(row0 + r) * C_DIM))[c4];
    float* dst = &xs[r * XS_STRIDE + c4 * 4];
    dst[0] = v.x; dst[1] = v.y; dst[2] = v.z; dst[3] = v.w;
  }
  __syncthreads();

  const int wave = tid >> 5;
  const int lane = tid & 31;
  const int half = lane >> 4;          // K sub-block selector (ISA 7.12.2)
  const int lm   = lane & 15;          // M (A) / N (B) within the 16x16 tile
  const int col  = wave * 16 + lm;     // 0..63, cols 60..63 hit zero-pad

  const float* arow = xs + lm * XS_STRIDE + 2 * half;
  const float* wrow = Wt + (size_t)col * C_DIM + 2 * half;

  v8f acc = {};
  for (int k0 = 0; k0 < C_DIM; k0 += 4) {
    v2f a;
    a.x = arow[k0];
    a.y = arow[k0 + 1];
    v2f bf;
    bf.x = wrow[k0];
    bf.y = wrow[k0 + 1];
    acc = __builtin_amdgcn_wmma_f32_16x16x4_f32(
        /*neg_a=*/false, a, /*neg_b=*/false, bf,
        /*c_mod=*/(short)0, acc, /*reuse_a=*/false, /*reuse_b=*/false);
  }

  if (col < NP) {
    const float bb = bias[col];
    // C/D layout: lane owns column N=lm, rows M = 8*half + r.
#pragma unroll
    for (int r = 0; r < 8; ++r) {
      const int m = 8 * half + r;
      params[(size_t)(row0 + m) * NP + col] = __expf(acc[r] + bb);
    }
  }
}

// -----------------------------------------------------------------------------
// Kernel 2: kappa = kappa_old + pre_kappa;  phi[b,l] (l=0..2048) scaled by
// L / text_len.  One block per batch row, 256 threads, 20 v_exp_f32 per l.
// -----------------------------------------------------------------------------
__global__ __launch_bounds__(256)
void phi_kernel(const float* __restrict__ params,
                const float* __restrict__ kappa_old,
                const int* __restrict__ text_lens,
                float* __restrict__ kappa_out,
                float* __restrict__ phi_out) {
  __shared__ float al[K_DIM];
  __shared__ float be[K_DIM];
  __shared__ float ka[K_DIM];
  const int b = blockIdx.x;
  const int tid = threadIdx.x;

  if (tid < NP) {
    const float p = params[(size_t)b * NP + tid];
    if (tid < K_DIM) {
      al[tid] = p;                               // alpha = cols 0..19
    } else if (tid < 2 * K_DIM) {
      be[tid - K_DIM] = p;                       // beta  = cols 20..39
    } else {
      const int k = tid - 2 * K_DIM;             // pre_kappa = cols 40..59
      const float kv = kappa_old[(size_t)b * K_DIM + k] + p;
      ka[k] = kv;
      kappa_out[(size_t)b * K_DIM + k] = kv;
    }
  }
  __syncthreads();

  const float scale = (float)L_DIM / (float)text_lens[b];
  for (int l = tid; l <= L_DIM; l += 256) {
    const float fl = (float)l;
    float acc = 0.0f;
#pragma unroll
    for (int k = 0; k < K_DIM; ++k) {
      const float d = ka[k] - fl;
      acc += al[k] * __expf(-be[k] * d * d);
    }
    phi_out[(size_t)b * (L_DIM + 1) + l] = acc * scale;
  }
}

// -----------------------------------------------------------------------------
// Kernel 3: w[b,v] = sum_l phi[b,l] * onehots[b,l,v].  One block (320 threads)
// per batch; one perfectly linear float4 stream over the 640KB slab (the
// 640MB total stream is the whole op's roofline: ~27.5us @ 23.3 TB/s).
// Thread t reads float4 index (t + 320*i); its row is exactly t/20 + 16*i.
// global_prefetch_b8 issued 16 iterations (~80KB) ahead to deepen the HBM stream.
// -----------------------------------------------------------------------------
__global__ __launch_bounds__(320)
void window_kernel(const float* __restrict__ phi,
                   const float* __restrict__ onehots,
                   float* __restrict__ w_out) {
  __shared__ float phi_s[L_DIM];
  __shared__ float red[320 * 4];
  const int b = blockIdx.x;
  const int tid = threadIdx.x;

  for (int i = tid; i < L_DIM; i += 320)
    phi_s[i] = phi[(size_t)b * (L_DIM + 1) + i];
  __syncthreads();

  const float4* O = (const float4*)(onehots + (size_t)b * L_DIM * V_DIM);
  const int lq = tid / 20;                       // row offset in 16-row stripe
  float ax = 0.0f, ay = 0.0f, az = 0.0f, aw = 0.0f;

  const int NITER = (L_DIM * V_DIM / 4) / 320;   // 128
  for (int i = 0; i < NITER; ++i) {
    if (i + 16 < NITER)
      __builtin_prefetch(&O[tid + 320 * (i + 16)], 0, 0);
    const float4 o = O[tid + 320 * i];
    const float p = phi_s[lq + 16 * i];
    ax += p * o.x; ay += p * o.y; az += p * o.z; aw += p * o.w;
  }

  red[tid * 4 + 0] = ax;
  red[tid * 4 + 1] = ay;
  red[tid * 4 + 2] = az;
  red[tid * 4 + 3] = aw;
  __syncthreads();

  if (tid < 20) {  // column-group c = tid covers v = 4c..4c+3
    float sx = 0.0f, sy = 0.0f, sz = 0.0f, sw = 0.0f;
    for (int g = 0; g < 16; ++g) {
      const int t = g * 20 + tid;
      sx += red[t * 4 + 0];
      sy += red[t * 4 + 1];
      sz += red[t * 4 + 2];
      sw += red[t * 4 + 3];
    }
    const size_t o = (size_t)b * V_DIM + tid * 4;
    w_out[o + 0] = sx;
    w_out[o + 1] = sy;
    w_out[o + 2] = sz;
    w_out[o + 3] = sw;
  }
}

// -----------------------------------------------------------------------------
extern "C" void kernel_launch(void* const* d_in, const int* in_sizes, int n_in,
                              void* d_out, int out_size, void* d_ws, size_t ws_size,
                              hipStream_t stream) {
  (void)in_sizes; (void)n_in; (void)out_size; (void)ws_size;
  const float* x         = (const float*)d_in[0];
  const float* W         = (const float*)d_in[1];
  const float* bias      = (const float*)d_in[2];
  const float* kappa_old = (const float*)d_in[3];
  const float* onehots   = (const float*)d_in[4];
  const int*   text_lens = (const int*)d_in[5];

  float* out       = (float*)d_out;
  float* w_out     = out;                                // [B, V]
  float* kappa_out = out + (size_t)B_DIM * V_DIM;        // [B, K]
  float* phi_out   = kappa_out + (size_t)B_DIM * K_DIM;  // [B, L+1]

  float* params_ws = (float*)d_ws;                       // [B, 60]
  float* Wt_ws     = params_ws + (size_t)B_DIM * NP;     // [64, 1024] padded W^T

  wpad_kernel<<<(NPAD * C_DIM) / 256, 256, 0, stream>>>(W, Wt_ws);
  params_gemm_kernel<<<B_DIM / 16, 128, 0, stream>>>(x, Wt_ws, bias, params_ws);
  phi_kernel<<<B_DIM, 256, 0, stream>>>(params_ws, kappa_old, text_lens,
                                        kappa_out, phi_out);
  window_kernel<<<B_DIM, 320, 0, stream>>>(phi_out, onehots, w_out);
}